// VolumeProfile_65369402245806
// MI455X (gfx1250) — compile-verified
//
#include <hip/hip_runtime.h>

// ---------------------------------------------------------------------------
// Volume profile: weighted histogram of 2^25 (price, volume) events.
//   Pass 1: global min/max of prices (wave32 butterfly + key atomics).
//   Pass 2: LDS-privatized weighted histogram, streamed via CDNA5 async
//           global->LDS DMA (ASYNCcnt), double-buffered, 2x b128 per chunk.
// Roofline: 2 x 256MB streaming reads @ 23.3 TB/s  ->  ~22 us floor.
// ---------------------------------------------------------------------------

__device__ __forceinline__ unsigned f2key(float f) {
    unsigned u = __float_as_uint(f);
    return (u & 0x80000000u) ? ~u : (u | 0x80000000u);   // order-preserving
}
__device__ __forceinline__ float key2f(unsigned k) {
    unsigned u = (k & 0x80000000u) ? (k & 0x7FFFFFFFu) : ~k;
    return __uint_as_float(u);
}

// ws layout: ws_u[0] = min key, ws_u[1] = max key.
// d_out (float[nb]) doubles as the global accumulation buffer.
__global__ void vp_init(unsigned* __restrict__ ws_u, float* __restrict__ out, int nb) {
    int i = blockIdx.x * blockDim.x + threadIdx.x;
    if (i == 0) { ws_u[0] = 0xFFFFFFFFu; ws_u[1] = 0u; }
    for (int b = i; b < nb; b += gridDim.x * blockDim.x) out[b] = 0.0f;
}

__global__ __launch_bounds__(256) void vp_minmax(const float4* __restrict__ in,
                                                 int nf4,
                                                 unsigned* __restrict__ ws_u) {
    const int stride = gridDim.x * blockDim.x;
    float lmin = 3.4e38f, lmax = -3.4e38f;
    for (int i = blockIdx.x * blockDim.x + threadIdx.x; i < nf4; i += stride) {
        float4 v = in[i];                 // (x,y)=(p0,v0), (z,w)=(p1,v1)
        lmin = fminf(lmin, fminf(v.x, v.z));
        lmax = fmaxf(lmax, fmaxf(v.x, v.z));
    }
    for (int off = 16; off > 0; off >>= 1) {          // wave32 butterfly
        lmin = fminf(lmin, __shfl_xor(lmin, off, 32));
        lmax = fmaxf(lmax, __shfl_xor(lmax, off, 32));
    }
    if ((threadIdx.x & 31) == 0) {
        atomicMin(&ws_u[0], f2key(lmin));
        atomicMax(&ws_u[1], f2key(lmax));
    }
}

__global__ __launch_bounds__(256) void vp_hist(const float4* __restrict__ in4,
                                               int nf4,
                                               const unsigned* __restrict__ ws_u,
                                               float* __restrict__ out, int nb) {
    __shared__ float  hist[1024];          // nb <= 1023 buckets (4 KB)
    __shared__ float4 stage[2][2][256];    // [buffer][sub][lane] = 16 KB staging
    const int tid = threadIdx.x;

    for (int b = tid; b < nb; b += blockDim.x) hist[b] = 0.0f;
    __syncthreads();

    const float pmin = key2f(ws_u[0]);
    const float pmax = key2f(ws_u[1]);
    // edges = linspace(pmin, pmax, nb+1): bucket width = range/nb
    const float inv = (pmax > pmin) ? (float)nb / (pmax - pmin) : 0.0f;
    const float hi  = (float)(nb - 1);

    const int tstride = gridDim.x * blockDim.x;               // 262144
    const int base    = blockIdx.x * blockDim.x + tid;
    const int span    = 2 * tstride;                          // f4s per chunk
    const int nchunks = (nf4 + span - 1) / span;

    // Issue one chunk = two coalesced b128 async loads per lane (32 B/lane).
    auto issue = [&](int c, int buf) {
        int ia = c * span + base;                // sub 0
        int ib = ia + tstride;                   // sub 1
        const float4* ga = in4 + (ia < nf4 ? ia : 0);
        const float4* gb = in4 + (ib < nf4 ? ib : 0);
        unsigned la = (unsigned)(size_t)(&stage[buf][0][tid]);
        unsigned lb = (unsigned)(size_t)(&stage[buf][1][tid]);
        asm volatile("global_load_async_to_lds_b128 %0, %1, off"
                     :: "v"(la), "v"(ga) : "memory");
        asm volatile("global_load_async_to_lds_b128 %0, %1, off"
                     :: "v"(lb), "v"(gb) : "memory");
    };
    auto bin2 = [&](float4 v, int gi) {
        if (gi < nf4) {
            int i0 = (int)fminf(fmaxf((v.x - pmin) * inv, 0.0f), hi);
            atomicAdd(&hist[i0], v.y);                         // ds_add_f32
            int i1 = (int)fminf(fmaxf((v.z - pmin) * inv, 0.0f), hi);
            atomicAdd(&hist[i1], v.w);
        }
    };

    if (nchunks > 0) issue(0, 0);
    for (int c = 0; c < nchunks; ++c) {
        int nc = (c + 1 < nchunks) ? c + 1 : c;               // last iter: dummy
        issue(nc, (c + 1) & 1);                               // ASYNCcnt -> 4
        asm volatile("s_wait_asynccnt 0x2" ::: "memory");     // chunk c landed
        float4 va = stage[c & 1][0][tid];
        float4 vb = stage[c & 1][1][tid];
        int ia = c * span + base;
        bin2(va, ia);
        bin2(vb, ia + tstride);
    }
    asm volatile("s_wait_asynccnt 0x0" ::: "memory");         // drain dummies
    __syncthreads();

    for (int b = tid; b < nb; b += blockDim.x)
        atomicAdd(&out[b], hist[b]);                          // one flush/bin/block
}

__global__ void vp_final(float* __restrict__ out, int nb) {
    int i = blockIdx.x * blockDim.x + threadIdx.x;
    if (i < nb) out[i] = truncf(out[i]);                      // int64 cast truncates
}

extern "C" void kernel_launch(void* const* d_in, const int* in_sizes, int n_in,
                              void* d_out, int out_size, void* d_ws, size_t ws_size,
                              hipStream_t stream) {
    (void)n_in; (void)ws_size;
    const float* in  = (const float*)d_in[0];   // [N,2] interleaved price,volume
    const int total  = in_sizes[0];             // 2*N floats (2^26)
    const int nf4    = total / 4;               // two events per float4 (2^24)
    const int nb     = out_size;                // bins-1 = 1023 buckets
    unsigned* ws_u   = (unsigned*)d_ws;         // 8 bytes: min/max keys
    float* out       = (float*)d_out;

    vp_init  <<<1, 256, 0, stream>>>(ws_u, out, nb);
    vp_minmax<<<2048, 256, 0, stream>>>((const float4*)in, nf4, ws_u);
    vp_hist  <<<1024, 256, 0, stream>>>((const float4*)in, nf4, ws_u, out, nb);
    vp_final <<<(nb + 255) / 256, 256, 0, stream>>>(out, nb);
}